// SOM_60215441490282
// MI455X (gfx1250) — compile-verified
//
#include <hip/hip_runtime.h>

typedef __attribute__((ext_vector_type(2))) float v2f;
typedef __attribute__((ext_vector_type(8))) float v8f;

#define MAP_M   256
#define MAP_N   256
#define DIMK    1024
#define UNITS   (MAP_M * MAP_N)
#define ROWS_PER_BLOCK 128   // 8 waves * 16 rows

// ---------------------------------------------------------------------------
// Kernel 0: initialize the packed argmin key in workspace to +inf/all-ones.
// ---------------------------------------------------------------------------
__global__ void som_init_ws(unsigned long long* ws) {
    ws[0] = 0xFFFFFFFFFFFFFFFFull;
}

// ---------------------------------------------------------------------------
// Kernel 1: BMU search via WMMA Gram-diagonal trick.
// Each wave owns 16 consecutive rows. For K-chunks of 4, the lane image of
// A = (W - x) in the 16x4 f32 A-layout is bit-identical to the 4x16 B-layout
// of A^T, so wmma(a, a, c) accumulates (D * D^T); its diagonal is the
// per-row squared distance, computed fully in f32 (RNE), matching the ref.
// ---------------------------------------------------------------------------
__global__ __launch_bounds__(256) void som_bmu_wmma(
    const float* __restrict__ W, const float* __restrict__ x,
    unsigned long long* __restrict__ ws) {

    __shared__ float xs[DIMK];
    for (int i = threadIdx.x; i < DIMK; i += 256) xs[i] = x[i];
    __syncthreads();

    const int lane  = threadIdx.x & 31;
    const int wave  = threadIdx.x >> 5;
    const int r0    = blockIdx.x * ROWS_PER_BLOCK + wave * 16;
    const int mrow  = lane & 15;        // matrix row handled by this lane
    const int khalf = (lane >> 4) << 1; // 0 for lanes 0-15, 2 for lanes 16-31

    const float* wp = W + (size_t)(r0 + mrow) * DIMK + khalf;

    v8f acc = {};
    #pragma unroll 8
    for (int k = 0; k < DIMK; k += 4) {
        // uniform-condition prefetch 256B ahead, every 128B of progress
        if ((k & 31) == 0) __builtin_prefetch(wp + 64, 0, 0);
        v2f wv = *(const v2f*)wp;                 // global_load_b64 (streamed)
        v2f xv = *(const v2f*)&xs[k + khalf];     // ds_load_b64
        v2f a;
        a.x = wv.x - xv.x;
        a.y = wv.y - xv.y;
        // D*D^T accumulate: same registers as A and B (see layout identity)
        acc = __builtin_amdgcn_wmma_f32_16x16x4_f32(
            /*neg_a=*/false, a, /*neg_b=*/false, a,
            /*c_mod=*/(short)0, acc, /*reuse_a=*/false, /*reuse_b=*/false);
        wp += 4;
    }

    // Diagonal extraction: C layout puts (M=r, N=lane) in VGPR r for lanes
    // 0-15 and (M=r+8, N=lane-16) for lanes 16-31.  Diagonal lives in
    // lanes 0-7 (rows 0-7, element index = lane) and lanes 24-31 (rows 8-15,
    // element index = lane-24).
    const int  sel   = (lane < 8) ? lane : (lane - 24);
    const bool valid = (lane < 8) || (lane >= 24);
    float v = acc[0];
    v = (sel == 1) ? acc[1] : v;
    v = (sel == 2) ? acc[2] : v;
    v = (sel == 3) ? acc[3] : v;
    v = (sel == 4) ? acc[4] : v;
    v = (sel == 5) ? acc[5] : v;
    v = (sel == 6) ? acc[6] : v;
    v = (sel == 7) ? acc[7] : v;

    float d  = valid ? v : __builtin_inff();
    int  row = r0 + ((lane < 8) ? lane : (lane - 16));

    // wave32 butterfly argmin; tie-break to smaller row (argmin-first).
    #pragma unroll
    for (int off = 16; off > 0; off >>= 1) {
        float od  = __shfl_xor(d,   off, 32);
        int   orw = __shfl_xor(row, off, 32);
        if (od < d || (od == d && orw < row)) { d = od; row = orw; }
    }

    if (lane == 0) {
        // d >= 0 so its IEEE bits are monotonic as u32; pack (dist, row)
        unsigned long long key =
            ((unsigned long long)__float_as_uint(d) << 32) | (unsigned)row;
        atomicMin(ws, key);   // global_atomic_min_u64
    }
}

// ---------------------------------------------------------------------------
// Kernel 2: streaming weight update, one row (1024 f32) per 256-thread block,
// b128 coalesced loads/stores.  out = w + lr*(x - w); inputs untouched.
// ---------------------------------------------------------------------------
__global__ __launch_bounds__(256) void som_update(
    const float* __restrict__ W, const float* __restrict__ x,
    const int* __restrict__ step, const unsigned long long* __restrict__ ws,
    float* __restrict__ out) {

    const int row = blockIdx.x;
    const unsigned bmu = (unsigned)(ws[0] & 0xFFFFFFFFull);

    // meshgrid(arange(M), arange(N)).ravel(): XC[p] = p % M, YC[p] = p / M
    const int bx = (int)(bmu & (MAP_M - 1));
    const int by = (int)(bmu >> 8);
    const int px = row & (MAP_M - 1);
    const int py = row >> 8;
    const float fdx = (float)(px - bx);
    const float fdy = (float)(py - by);
    const float g2  = fdx * fdx + fdy * fdy;

    const float dec     = powf(0.999f, (float)step[0]);
    const float alpha   = 0.3f * dec;
    const float sigma   = 128.0f * dec;            // max(M,N)/2 * decay^step
    const float lr      = alpha * expf(-g2 / (2.0f * sigma * sigma));

    const int c = threadIdx.x << 2;                // 4 floats per thread
    const float4 wv = *(const float4*)(W + (size_t)row * DIMK + c);
    const float4 xv = *(const float4*)(x + c);
    float4 o;
    o.x = wv.x + lr * (xv.x - wv.x);
    o.y = wv.y + lr * (xv.y - wv.y);
    o.z = wv.z + lr * (xv.z - wv.z);
    o.w = wv.w + lr * (xv.w - wv.w);
    *(float4*)(out + (size_t)row * DIMK + c) = o;
}

// ---------------------------------------------------------------------------
extern "C" void kernel_launch(void* const* d_in, const int* in_sizes, int n_in,
                              void* d_out, int out_size, void* d_ws, size_t ws_size,
                              hipStream_t stream) {
    (void)in_sizes; (void)n_in; (void)out_size; (void)ws_size;
    const float* x  = (const float*)d_in[0];
    const float* W  = (const float*)d_in[1];
    const int* step = (const int*)d_in[2];
    float* out = (float*)d_out;
    unsigned long long* ws = (unsigned long long*)d_ws;

    som_init_ws<<<1, 1, 0, stream>>>(ws);
    som_bmu_wmma<<<UNITS / ROWS_PER_BLOCK, 256, 0, stream>>>(W, x, ws);
    som_update<<<UNITS, 256, 0, stream>>>(W, x, step, ws, out);
}